// BernoulliMaskedPPCA_12094627905903
// MI455X (gfx1250) — compile-verified
//
#include <hip/hip_runtime.h>
#include <hip/hip_bf16.h>
#include <math.h>

// ---------------------------------------------------------------------------
// BernoulliMaskedPPCA, collapsed form:
//   u0,u1,v = x @ [M0 M1 Mb]   (dense masked 65536x784 @ 784x3, f32 WMMA)
//   out = -(D/(P*m*N)) * sum_n ( log_w + v_n + logsumexp_j( g[j] + u0*z0[j] + u1*z1[j] ) )
// Memory-bound: 205 MB of x @ 23.3 TB/s ~ 9 us floor.
// ---------------------------------------------------------------------------

typedef float v2f __attribute__((ext_vector_type(2)));
typedef float v8f __attribute__((ext_vector_type(8)));

#define N_OBS   65536
#define D_DIM   784
#define M_DIM   117          // int(784*0.15)
#define L_BINS  20
#define G_PTS   400          // L^2
#define KCHUNKS 196          // 784/4
#define ROWS_PER_BLOCK 128   // 8 waves * 16 rows
#define LOG_W   (-1.3862943611198906f)   // 2*log(10/20)
#define NEG_LOG_2PI (-1.8378770664093453f)
#define COEF    (784.0f / (4.0f * 117.0f * 65536.0f))

// ws layout (floats)
#define WS_M0   0
#define WS_M1   784
#define WS_MB   1568
#define WS_B    2352                  // 196*64 = 12544 floats, WMMA B layout
#define WS_Z0   (WS_B + 196*64)      // 14896
#define WS_Z1   (WS_Z0 + 400)
#define WS_G    (WS_Z1 + 400)        // end: 16096 floats = 64,384 B

// ---------------------------------------------------------------------------
// Kernel 1 (1 block, 256 threads): build masked weight vectors, the WMMA-laid-
// out B table, the 400-point grid tables g/z0/z1, and zero d_out.
// ---------------------------------------------------------------------------
__global__ void ppca_setup_kernel(const float* __restrict__ W,
                                  const float* __restrict__ b,
                                  const int*   __restrict__ perms,
                                  float* __restrict__ ws,
                                  float* __restrict__ out) {
    const int tid = threadIdx.x;
    if (tid == 0) out[0] = 0.0f;

    float* M0 = ws + WS_M0;
    float* M1 = ws + WS_M1;
    float* Mb = ws + WS_MB;
    float* Bt = ws + WS_B;

    for (int d = tid; d < D_DIM; d += 256) { M0[d] = 0.f; M1[d] = 0.f; Mb[d] = 0.f; }
    __syncthreads();

    const int* perm = perms + 3 * D_DIM;     // only the LAST permutation matters
    for (int i = tid; i < M_DIM; i += 256) {
        int d = perm[i];
        M0[d] = W[2 * d + 0];
        M1[d] = W[2 * d + 1];
        Mb[d] = b[d];
    }
    __syncthreads();

    // B table in V_WMMA_F32_16X16X4 B layout, paired per lane for b64 loads:
    // Bt[c*64 + lane*2 + vgpr] = B[K = 4c + (lane>=16 ? 2:0) + vgpr][N = lane%16]
    for (int idx = tid; idx < KCHUNKS * 64; idx += 256) {
        int c    = idx >> 6;
        int q    = idx & 63;
        int lane = q >> 1;
        int vg   = q & 1;
        int k    = 4 * c + ((lane >= 16) ? 2 : 0) + vg;
        int n    = lane & 15;
        float val = 0.f;
        if      (n == 0) val = M0[k];
        else if (n == 1) val = M1[k];
        else if (n == 2) val = Mb[k];
        Bt[idx] = val;
    }

    // grid tables: z (meshgrid 'xy', row-major flatten) and
    // g[j] = log_p_z[j] + sum_i log_sigmoid(-logits[i][j])
    for (int j = tid; j < G_PTS; j += 256) {
        float zx0 = -5.0f + (float)(j % L_BINS) * (10.0f / 19.0f);
        float zx1 = -5.0f + (float)(j / L_BINS) * (10.0f / 19.0f);
        float C = 0.0f;
        for (int i = 0; i < M_DIM; ++i) {
            int d = perm[i];
            float lg = fmaf(W[2 * d], zx0, fmaf(W[2 * d + 1], zx1, b[d]));
            // log_sigmoid(-lg) = min(-lg, 0) - log1p(exp(-|lg|))
            C += fminf(-lg, 0.0f) - log1pf(__expf(-fabsf(lg)));
        }
        ws[WS_Z0 + j] = zx0;
        ws[WS_Z1 + j] = zx1;
        ws[WS_G  + j] = NEG_LOG_2PI - 0.5f * (zx0 * zx0 + zx1 * zx1) + C;
    }
}

// ---------------------------------------------------------------------------
// Kernel 2 (512 blocks x 256 threads): streaming f32 WMMA GEMM (u0,u1,v) fused
// with per-row 400-point logsumexp and a block-level reduction -> atomicAdd.
// ---------------------------------------------------------------------------
__global__ void __launch_bounds__(256)
ppca_main_kernel(const float* __restrict__ x,
                 const float* __restrict__ ws,
                 float* __restrict__ out) {
    __shared__ float s_z0[G_PTS], s_z1[G_PTS], s_g[G_PTS];
    __shared__ float s_uv[ROWS_PER_BLOCK * 3];
    __shared__ float s_m[256], s_s[256];

    const int tid   = threadIdx.x;
    const int wave  = tid >> 5;
    const int lane  = tid & 31;
    const int laneN = lane & 15;              // N column of B/D; row M for A loads
    const int halfK = (lane >> 4) * 2;        // 0 for lanes 0-15, 2 for lanes 16-31

    // stage grid tables into LDS (consumed after the next barrier)
    for (int j = tid; j < G_PTS; j += 256) {
        s_z0[j] = ws[WS_Z0 + j];
        s_z1[j] = ws[WS_Z1 + j];
        s_g[j]  = ws[WS_G  + j];
    }

    // ---- Phase A: u0,u1,v via V_WMMA_F32_16X16X4_F32 ------------------------
    const long rowbase = (long)blockIdx.x * ROWS_PER_BLOCK + wave * 16;
    const float* __restrict__ xrow = x + (rowbase + laneN) * (long)D_DIM + halfK;
    const float* __restrict__ Bt   = ws + WS_B;

    v8f acc = {};
    for (int c = 0; c < KCHUNKS; ++c) {
        v2f a  = *(const v2f*)(xrow + 4 * c);            // A: lane holds K pair of its row
        v2f bb = *(const v2f*)(Bt + c * 64 + lane * 2);  // B: precomputed WMMA layout
        acc = __builtin_amdgcn_wmma_f32_16x16x4_f32(
            /*neg_a=*/false, a, /*neg_b=*/false, bb,
            /*c_mod=*/(short)0, acc, /*reuse_a=*/false, /*reuse_b=*/false);
    }

    // D layout: VGPR r -> M = r + 8*(lane/16), N = lane%16. Keep N<3 only.
    if (laneN < 3) {
        int mbase = wave * 16 + (lane >> 4) * 8;
        #pragma unroll
        for (int r = 0; r < 8; ++r)
            s_uv[(mbase + r) * 3 + laneN] = acc[r];
    }
    __syncthreads();

    // ---- Phase B: per-row logsumexp over 400 grid points --------------------
    const int row = tid & 127;                // row within block
    const int h   = tid >> 7;                 // half of the grid (0..1)
    const float u0 = s_uv[row * 3 + 0];
    const float u1 = s_uv[row * 3 + 1];
    const float vv = s_uv[row * 3 + 2];

    const int j0 = h * (G_PTS / 2);
    float mmax = -INFINITY;
    for (int j = j0; j < j0 + G_PTS / 2; ++j) {
        float t = fmaf(u0, s_z0[j], fmaf(u1, s_z1[j], s_g[j]));
        mmax = fmaxf(mmax, t);
    }
    float ssum = 0.0f;
    for (int j = j0; j < j0 + G_PTS / 2; ++j) {
        float t = fmaf(u0, s_z0[j], fmaf(u1, s_z1[j], s_g[j]));
        ssum += __expf(t - mmax);
    }
    s_m[tid] = mmax;
    s_s[tid] = ssum;
    __syncthreads();

    float rowval = 0.0f;
    if (h == 0) {
        float mA = mmax, sA = ssum;
        float mB = s_m[row + 128], sB = s_s[row + 128];
        float mm = fmaxf(mA, mB);
        float ss = sA * __expf(mA - mm) + sB * __expf(mB - mm);
        rowval = LOG_W + vv + mm + __logf(ss);
    }
    __syncthreads();                          // s_m reuse
    s_m[tid] = rowval;
    __syncthreads();
    for (int off = 128; off > 0; off >>= 1) {
        if (tid < off) s_m[tid] += s_m[tid + off];
        __syncthreads();
    }
    if (tid == 0) atomicAdd(out, -COEF * s_m[0]);
}

// ---------------------------------------------------------------------------
extern "C" void kernel_launch(void* const* d_in, const int* in_sizes, int n_in,
                              void* d_out, int out_size, void* d_ws, size_t ws_size,
                              hipStream_t stream) {
    const float* x     = (const float*)d_in[0];   // (65536, 784)
    const float* W     = (const float*)d_in[1];   // (784, 2)
    const float* b     = (const float*)d_in[2];   // (784,)
    const int*   perms = (const int*)d_in[3];     // (4, 784)
    // d_in[4] = bins (=20), fixed at compile time above
    float* out = (float*)d_out;
    float* ws  = (float*)d_ws;                    // needs 64,384 bytes

    ppca_setup_kernel<<<1, 256, 0, stream>>>(W, b, perms, ws, out);
    ppca_main_kernel<<<N_OBS / ROWS_PER_BLOCK, 256, 0, stream>>>(x, ws, out);
}